// FrequencyLayer_83966610636992
// MI455X (gfx1250) — compile-verified
//
#include <hip/hip_runtime.h>
#include <math.h>

// ---------------------------------------------------------------------------
// BSARec FrequencyLayer, CUTOFF=2  ==>  low-pass = DC + first harmonic only.
//   low[b,s,h] = (S0 + 2*(C*cos(2*pi*s/S) + D*sin(2*pi*s/S))) / S
//   y = x*(1+beta^2) + low*(1-beta^2);  out = LayerNorm_H(y)
// Pass1 (column sums) is a real GEMM: R[3,H] = W[3,S] x X[S,H] per batch,
// mapped onto V_WMMA_F32_16X16X4_F32 (fp32 accumulate, no precision loss).
// Memory-bound: ~384 MiB total traffic -> ~16 us @ 23.3 TB/s HBM.
// Inner loop: 1 ds_load_b64 (A tile) + 2 global_load_b32 (B tile) + 1 wmma
// per 256 B of x -> instruction-lean enough to saturate HBM.
// ---------------------------------------------------------------------------

typedef __attribute__((ext_vector_type(2))) float v2f;
typedef __attribute__((ext_vector_type(8))) float v8f;

#define BB   64
#define SS   2048
#define HH   256
#define SCH  8            // s-chunks for pass-1 parallelism
#define SPC  (SS / SCH)   // 256 s-values per chunk
#define NSTEP (SPC / 4)   // 64 WMMA steps of K=4

// workspace layout (float offsets)
#define WS_CT   0
#define WS_ST   (WS_CT + SS)
#define WS_PART (WS_ST + SS)                  // [SCH][B][3][H]
#define WS_COL  (WS_PART + SCH * BB * 3 * HH) // [B][3][H]

// ---------------------------------------------------------------- trig table
__global__ void k_table(float* __restrict__ ct, float* __restrict__ st) {
  int t = blockIdx.x * blockDim.x + threadIdx.x;
  if (t < SS) {
    float ang = (float)(6.283185307179586 * (double)t / (double)SS);
    ct[t] = cosf(ang);
    st[t] = sinf(ang);
  }
}

// ------------------------------------------------- pass 1: WMMA column sums
// grid: BB * SCH * 2 blocks, 256 thr (8 waves). Wave handles one 16-wide
// h-tile over 256 s-values. A = weight tile {1,cos,sin,0...}, B = x tile.
// A layout (16x4 f32): lane L, vgpr r -> M = L&15, K = 2*(L>>4) + r.
// B layout (4x16 f32): lane L, vgpr r -> N = L&15, K = 2*(L>>4) + r.
// Per-lane A values come from a fixed row of a 4x256 LDS table
// {ones, cos, sin, zeros} -> a single 8B-aligned ds_load_b64 per step,
// no per-step selects.
__global__ __launch_bounds__(256) void k_colsum_wmma(
    const float* __restrict__ x, float* __restrict__ part,
    const float* __restrict__ ct, const float* __restrict__ st) {
  __shared__ float awtab[4 * SPC];   // row0=1.0, row1=cos, row2=sin, row3=0.0

  int bid   = blockIdx.x;
  int hhalf = bid & 1;
  int sch   = (bid >> 1) & (SCH - 1);
  int b     = bid >> 4;
  int s0    = sch * SPC;

  int tid = threadIdx.x;
  awtab[0 * SPC + tid] = 1.0f;
  awtab[1 * SPC + tid] = ct[s0 + tid];
  awtab[2 * SPC + tid] = st[s0 + tid];
  awtab[3 * SPC + tid] = 0.0f;
  __syncthreads();

  int wave = tid >> 5, lane = tid & 31;
  int h0    = (hhalf * 8 + wave) * 16;
  int n     = lane & 15;     // M for A, N for B (same index math)
  int khalf = lane >> 4;

  int arow = (n < 3) ? n : 3;                    // 0:ones 1:cos 2:sin 3:zeros
  const float* ap = awtab + arow * SPC + 2 * khalf;
  const float* px = x + ((size_t)b * SS + s0 + 2 * khalf) * HH + h0 + n;

  v8f acc = {};

#pragma unroll 8
  for (int step = 0; step < NSTEP; ++step) {
    v2f av = *(const v2f*)(ap + step * 4);        // A[k], A[k+1] (8B aligned)
    float b0 = px[(size_t)step * 4 * HH];
    float b1 = px[(size_t)step * 4 * HH + HH];
    v2f bv = {b0, b1};
    acc = __builtin_amdgcn_wmma_f32_16x16x4_f32(
        /*neg_a=*/false, av, /*neg_b=*/false, bv,
        /*c_mod=*/(short)0, acc, /*reuse_a=*/false, /*reuse_b=*/false);
  }

  // C layout: vgpr r, lanes 0-15 hold row M=r. Rows 0..2 = {S0, C, D}.
  if (lane < 16) {
    float* pp = part + (((size_t)sch * BB + b) * 3) * HH + h0 + lane;
    pp[0 * HH] = acc[0];
    pp[1 * HH] = acc[1];
    pp[2 * HH] = acc[2];
  }
}

// ------------------------------------- pass 1b: deterministic chunk combine
__global__ void k_reduce(const float* __restrict__ part, float* __restrict__ col) {
  int t = blockIdx.x * 256 + threadIdx.x;   // 0 .. B*H-1
  int b = t >> 8, h = t & 255;
#pragma unroll
  for (int w = 0; w < 3; ++w) {
    float s = 0.0f;
    for (int c = 0; c < SCH; ++c)
      s += part[(((size_t)c * BB + b) * 3 + w) * HH + h];
    col[((size_t)b * 3 + w) * HH + h] = s;
  }
}

// --------------------------------- pass 2: reconstruct + residual + LayerNorm
// One wave per (b,s) row; lane owns 8 channels (two float4 loads).
__global__ __launch_bounds__(256) void k_apply_ln(
    const float* __restrict__ x, const float* __restrict__ sqb,
    const float* __restrict__ lnw, const float* __restrict__ lnb,
    const float* __restrict__ col, const float* __restrict__ ct,
    const float* __restrict__ st, float* __restrict__ out) {
  int wave = threadIdx.x >> 5, lane = threadIdx.x & 31;
  long row = (long)blockIdx.x * 8 + wave;
  int b = (int)(row >> 11);
  int s = (int)(row & (SS - 1));

  const float4* xr = (const float4*)(x + (size_t)row * HH);
  const float4* qr = (const float4*)sqb;
  const float4* wr = (const float4*)lnw;
  const float4* br = (const float4*)lnb;
  const float4* s0r = (const float4*)(col + (size_t)b * 3 * HH);
  const float4* c1r = (const float4*)(col + (size_t)b * 3 * HH + HH);
  const float4* d1r = (const float4*)(col + (size_t)b * 3 * HH + 2 * HH);

  float4 xv[2]  = {xr[lane * 2],  xr[lane * 2 + 1]};
  float4 qv[2]  = {qr[lane * 2],  qr[lane * 2 + 1]};
  float4 s0v[2] = {s0r[lane * 2], s0r[lane * 2 + 1]};
  float4 c1v[2] = {c1r[lane * 2], c1r[lane * 2 + 1]};
  float4 d1v[2] = {d1r[lane * 2], d1r[lane * 2 + 1]};

  float cc = ct[s], dd = st[s];
  const float invS = 1.0f / (float)SS;

  float yv[8];
  float ysum = 0.0f, ysq = 0.0f;
#pragma unroll
  for (int j = 0; j < 8; ++j) {
    float xj = ((const float*)xv)[j];
    float qj = ((const float*)qv)[j];
    float lp = (((const float*)s0v)[j] +
                2.0f * (((const float*)c1v)[j] * cc +
                        ((const float*)d1v)[j] * dd)) * invS;
    float b2 = qj * qj;
    float y  = xj * (1.0f + b2) + lp * (1.0f - b2);
    yv[j] = y;
    ysum += y;
    ysq  += y * y;
  }

  // wave32 tree reduction over the row's 256 values
#pragma unroll
  for (int off = 16; off > 0; off >>= 1) {
    ysum += __shfl_xor(ysum, off);
    ysq  += __shfl_xor(ysq,  off);
  }
  float mean = ysum * (1.0f / (float)HH);
  float var  = ysq  * (1.0f / (float)HH) - mean * mean;
  float inv  = rsqrtf(var + 1e-5f);

  float4 ov[2];
  float4 wv[2] = {wr[lane * 2], wr[lane * 2 + 1]};
  float4 bv[2] = {br[lane * 2], br[lane * 2 + 1]};
#pragma unroll
  for (int j = 0; j < 8; ++j) {
    ((float*)ov)[j] = (yv[j] - mean) * inv * ((const float*)wv)[j] +
                      ((const float*)bv)[j];
  }
  float4* orow = (float4*)(out + (size_t)row * HH);
  orow[lane * 2]     = ov[0];
  orow[lane * 2 + 1] = ov[1];
}

// ---------------------------------------------------------------------------
extern "C" void kernel_launch(void* const* d_in, const int* in_sizes, int n_in,
                              void* d_out, int out_size, void* d_ws, size_t ws_size,
                              hipStream_t stream) {
  const float* x   = (const float*)d_in[0];
  const float* sqb = (const float*)d_in[1];
  const float* lnw = (const float*)d_in[2];
  const float* lnb = (const float*)d_in[3];
  float* out = (float*)d_out;
  float* ws  = (float*)d_ws;

  float* ct   = ws + WS_CT;
  float* st   = ws + WS_ST;
  float* part = ws + WS_PART;
  float* col  = ws + WS_COL;

  k_table<<<SS / 256, 256, 0, stream>>>(ct, st);
  k_colsum_wmma<<<BB * SCH * 2, 256, 0, stream>>>(x, part, ct, st);
  k_reduce<<<BB * HH / 256, 256, 0, stream>>>(part, col);
  k_apply_ln<<<BB * SS / 8, 256, 0, stream>>>(x, sqb, lnw, lnb, col, ct, st, out);
}